// LatentAggregator_26061861552498
// MI455X (gfx1250) — compile-verified
//
#include <hip/hip_runtime.h>
#include <hip/hip_bf16.h>

typedef __attribute__((ext_vector_type(2))) float v2f;
typedef __attribute__((ext_vector_type(8))) float v8f;

#define NL    9
#define NB    4
#define NQ    300
#define NC    256
#define NCCLS 81
#define NS    100
#define SPAD  112
#define NST   7        // s-tiles (SPAD/16)
#define NHW   25600
#define LQ    2700     // NL*NQ
#define QPAD  2704     // 169*16
#define EPSV  1e-6f

static __device__ __forceinline__ v8f wmma4(v2f a, v2f b, v8f c) {
  // D = A(16x4 f32) * B(4x16 f32) + C(16x16 f32)
  return __builtin_amdgcn_wmma_f32_16x16x4_f32(false, a, false, b, (short)0, c,
                                               false, false);
}

// ---------------------------------------------------------------------------
// Kernel 1: W[b][s][q] = relu( Qsig[b,q,:] . Seed[b,s,:] ) * valid[b,s]
// One wave per (b, q-tile); 7 s-tile accumulators share the A fragment.
// Branch-free inner loop: OOB rows clamped, zeroed at store.
// Wt stored s-major (transposed) with padded q rows written as exact zeros.
// grid = (169, B), block = 32
// ---------------------------------------------------------------------------
__global__ void la_sim_relu_kernel(const float* __restrict__ q_sig,
                                   const float* __restrict__ seed,
                                   const unsigned char* __restrict__ pad_mask,
                                   float* __restrict__ Wt) {
  const int b    = blockIdx.y;
  const int qt   = blockIdx.x;
  const int lane = threadIdx.x;
  const int half = lane >> 4;
  const int m    = lane & 15;

  // A fragment row (q-dimension), clamped
  const int qrowA = min(qt * 16 + m, LQ - 1);
  const int l = qrowA / NQ, qq = qrowA % NQ;       // [L,B,Q,C] layout
  const float* aptr = q_sig + (((size_t)l * NB + b) * NQ + qq) * NC;

  const float* sbase = seed + (size_t)b * NS * NC;
  const float* brow[NST];
#pragma unroll
  for (int st = 0; st < NST; ++st)
    brow[st] = sbase + (size_t)min(st * 16 + m, NS - 1) * NC;

  v8f acc[NST] = {};
#pragma unroll 4
  for (int k0 = 0; k0 < NC; k0 += 4) {
    const v2f a = *(const v2f*)(aptr + k0 + 2 * half);
#pragma unroll
    for (int st = 0; st < NST; ++st) {
      const v2f bb = *(const v2f*)(brow[st] + k0 + 2 * half);
      acc[st] = wmma4(a, bb, acc[st]);
    }
  }

  // C/D layout: N = lane%16 (s), M = v + 8*half (q)
#pragma unroll
  for (int st = 0; st < NST; ++st) {
    const int sOut = st * 16 + m;                  // < SPAD
    const float vmask =
        (sOut < NS && pad_mask[b * NS + sOut] != 0) ? 1.f : 0.f;
#pragma unroll
    for (int v = 0; v < 8; ++v) {
      const int qrow = qt * 16 + v + 8 * half;     // < QPAD
      const float x = (qrow < LQ) ? fmaxf(acc[st][v], 0.f) * vmask : 0.f;
      Wt[((size_t)b * SPAD + sOut) * QPAD + qrow] = x;
    }
  }
}

// ---------------------------------------------------------------------------
// Kernel 2: rinv[b][s] = valid[b,s] / max(sum_q W[b][s][q], eps)
// grid = B*SPAD blocks, block = 1 wave
// ---------------------------------------------------------------------------
__global__ void la_denom_kernel(const float* __restrict__ Wt,
                                const unsigned char* __restrict__ pad_mask,
                                float* __restrict__ rinv) {
  const int idx = blockIdx.x;                      // 0 .. NB*SPAD-1
  const int b = idx / SPAD, s = idx % SPAD;
  const float* row = Wt + ((size_t)b * SPAD + s) * QPAD;
  float sum = 0.f;
  for (int q = threadIdx.x; q < QPAD; q += 32) sum += row[q];
  for (int off = 16; off > 0; off >>= 1) sum += __shfl_down(sum, off, 32);
  if (threadIdx.x == 0) {
    const float vmask = (s < NS && pad_mask[b * NS + s] != 0) ? 1.f : 0.f;
    rinv[idx] = vmask / fmaxf(sum, EPSV);
  }
}

// ---------------------------------------------------------------------------
// Kernel 3: out[b][s][c] = rinv[b][s] * sum_q W[b][s][q] * Qx[b,q,c]
// One wave per (b, c-tile); 7 s-tile accumulators share the B fragment.
// K loop structured per layer (300 = 75*4) -> no division, no q guards.
// Padded Wt rows (q>=2700) are never touched.
// grid = (ceil(NDIM/16), B), block = 32
// ---------------------------------------------------------------------------
template <int NDIM>
__global__ void la_proto_kernel(const float* __restrict__ Wt,
                                const float* __restrict__ qx,   // [L,B,Q,NDIM]
                                const float* __restrict__ rinv,
                                float* __restrict__ out) {      // [B,NS,NDIM]
  const int b    = blockIdx.y;
  const int ct   = blockIdx.x;
  const int lane = threadIdx.x;
  const int half = lane >> 4;
  const int m    = lane & 15;

  const int ccolB = min(ct * 16 + m, NDIM - 1);    // clamped B column
  const float* wbase = Wt + ((size_t)b * SPAD + m) * QPAD;  // + st*16*QPAD

  v8f acc[NST] = {};
  for (int l = 0; l < NL; ++l) {
    const float* bbase = qx + (((size_t)l * NB + b) * NQ) * NDIM + ccolB;
    const int kbase = l * NQ;
#pragma unroll 4
    for (int kq = 0; kq < NQ; kq += 4) {
      const int kin = kq + 2 * half;
      v2f bb;
      bb.x = bbase[(size_t)kin * NDIM];
      bb.y = bbase[(size_t)(kin + 1) * NDIM];
      const int k = kbase + kin;
#pragma unroll
      for (int st = 0; st < NST; ++st) {
        const v2f a = *(const v2f*)(wbase + (size_t)st * 16 * QPAD + k);
        acc[st] = wmma4(a, bb, acc[st]);
      }
    }
  }

  const int cOut = ct * 16 + m;
#pragma unroll
  for (int st = 0; st < NST; ++st) {
#pragma unroll
    for (int v = 0; v < 8; ++v) {
      const int srow = st * 16 + v + 8 * half;
      if (srow < NS && cOut < NDIM)
        out[((size_t)b * NS + srow) * NDIM + cOut] =
            acc[st][v] * rinv[b * SPAD + srow];
    }
  }
}

// ---------------------------------------------------------------------------
// Kernel 4: proto_masks[b][s][hw] = P_emb[b,s,:] . mask_features[b,:,hw]
// One wave per (b, hw-tile); 7 s-tile accumulators share the B fragment
// (cuts mask_features fragment loads 7x). P_emb already carries valid mask.
// grid = (1600, B), block = 32
// ---------------------------------------------------------------------------
__global__ void la_masks_kernel(const float* __restrict__ pemb,   // [B,NS,NC]
                                const float* __restrict__ mfeat,  // [B,NC,NHW]
                                float* __restrict__ out) {        // [B,NS,NHW]
  const int b    = blockIdx.y;
  const int nt   = blockIdx.x;
  const int lane = threadIdx.x;
  const int half = lane >> 4;
  const int m    = lane & 15;

  const float* abase = pemb + (size_t)b * NS * NC;
  const float* arow[NST];
#pragma unroll
  for (int st = 0; st < NST; ++st)
    arow[st] = abase + (size_t)min(st * 16 + m, NS - 1) * NC;

  const float* bbase = mfeat + (size_t)b * NC * NHW + nt * 16 + m;

  v8f acc[NST] = {};
#pragma unroll 4
  for (int k0 = 0; k0 < NC; k0 += 4) {
    const int k = k0 + 2 * half;
    v2f bb;
    bb.x = bbase[(size_t)k * NHW];
    bb.y = bbase[(size_t)(k + 1) * NHW];
#pragma unroll
    for (int st = 0; st < NST; ++st) {
      const v2f a = *(const v2f*)(arow[st] + k);
      acc[st] = wmma4(a, bb, acc[st]);
    }
  }

  const int hwOut = nt * 16 + m;                   // NHW = 1600*16, exact
#pragma unroll
  for (int st = 0; st < NST; ++st) {
#pragma unroll
    for (int v = 0; v < 8; ++v) {
      const int srow = st * 16 + v + 8 * half;
      if (srow < NS)
        out[((size_t)b * NS + srow) * NHW + hwOut] = acc[st][v];
    }
  }
}

// ---------------------------------------------------------------------------
extern "C" void kernel_launch(void* const* d_in, const int* in_sizes, int n_in,
                              void* d_out, int out_size, void* d_ws,
                              size_t ws_size, hipStream_t stream) {
  (void)in_sizes; (void)n_in; (void)out_size; (void)ws_size;

  const float* q_cls = (const float*)d_in[0];  // [L,B,Q,Ccls]
  const float* q_emb = (const float*)d_in[1];  // [L,B,Q,C]
  const float* q_sig = (const float*)d_in[2];  // [L,B,Q,C]
  const float* seed  = (const float*)d_in[3];  // [B,S,C]
  const float* mfeat = (const float*)d_in[4];  // [B,C,H,W]
  const unsigned char* pad = (const unsigned char*)d_in[5];  // [B,S] bool

  float* out     = (float*)d_out;
  float* out_cls = out;                                   // 4*100*81
  float* out_msk = out + (size_t)NB * NS * NCCLS;         // 4*100*25600
  float* out_emb = out_msk + (size_t)NB * NS * NHW;       // 4*100*256

  float* Wt   = (float*)d_ws;                             // [B][SPAD][QPAD]
  float* rinv = Wt + (size_t)NB * SPAD * QPAD;            // [B][SPAD]

  const dim3 wave(32, 1, 1);
  la_sim_relu_kernel<<<dim3(QPAD / 16, NB), wave, 0, stream>>>(q_sig, seed,
                                                              pad, Wt);
  la_denom_kernel<<<dim3(NB * SPAD, 1, 1), wave, 0, stream>>>(Wt, pad, rinv);
  la_proto_kernel<NC><<<dim3(NC / 16, NB), wave, 0, stream>>>(Wt, q_emb, rinv,
                                                              out_emb);
  la_proto_kernel<NCCLS><<<dim3(6, NB), wave, 0, stream>>>(Wt, q_cls, rinv,
                                                           out_cls);
  la_masks_kernel<<<dim3(NHW / 16, NB), wave, 0, stream>>>(out_emb, mfeat,
                                                           out_msk);
}